// ANRB_Conv_50929722196812
// MI455X (gfx1250) — compile-verified
//
#include <hip/hip_runtime.h>
#include <hip/hip_bf16.h>

// ---------------------------------------------------------------------------
// Problem constants (from reference): B=2, C=64, H=W=96, K=7
//   HP=WP=90, N=8100, KFLAT=C*K*K=3136
// ---------------------------------------------------------------------------
#define HB      2
#define HC      64
#define HH      96
#define HW      96
#define HHP     90
#define HN      8100          // 90*90
#define HROWS   16200         // B*N
#define KFLAT   3136          // 64*7*7
#define KITERS  784           // KFLAT/4
#define XPLANE  9216          // 96*96
#define XBATCH  589824        // 64*96*96
#define XTOTAL  1179648       // 2*64*96*96

typedef __attribute__((ext_vector_type(2))) float v2f;
typedef __attribute__((ext_vector_type(8))) float v8f;

// ---------------------------------------------------------------------------
// Kernel 1: fused QKV 7x7 valid conv as im2col GEMM on the fp32 WMMA path.
//   One wave (32 threads) per tile of 16 output pixels. A = 16x4 fp32 im2col
//   slice, B = 4x16 fp32 weight slice (cols 0..2 = wq/wk/wv, rest 0),
//   accumulate 16x16 f32 with V_WMMA_F32_16X16X4_F32, 784 iterations.
//   Weights (packed [k][4]) and the im2col address-offset table live in LDS.
// ---------------------------------------------------------------------------
__global__ void __launch_bounds__(32)
qkv_conv_wmma(const float* __restrict__ x,
              const float* __restrict__ wq,
              const float* __restrict__ wk,
              const float* __restrict__ wv,
              const float* __restrict__ bq,
              const float* __restrict__ bk,
              const float* __restrict__ bv,
              float* __restrict__ qo,
              float* __restrict__ ko,
              float* __restrict__ vo)
{
    __shared__ float sW[KFLAT * 4];   // [k][n], n=0:wq 1:wk 2:wv 3:pad  (49KB)
    __shared__ int   sOff[KFLAT];     // im2col element offset per k      (12.25KB)

    const int lane = threadIdx.x;     // wave32

    for (int k = lane; k < KFLAT; k += 32) {
        sW[k * 4 + 0] = wq[k];
        sW[k * 4 + 1] = wk[k];
        sW[k * 4 + 2] = wv[k];
        sW[k * 4 + 3] = 0.0f;
        int c = k / 49, rem = k % 49;
        int r = rem / 7, s = rem % 7;
        sOff[k] = c * XPLANE + r * HW + s;
    }
    __syncthreads();

    const int half = lane >> 4;       // 0/1
    const int m    = lane & 15;       // row for A, column N for B/C/D
    const int n    = m;

    // pixel handled by this lane's A row
    long p = (long)blockIdx.x * 16 + m;
    if (p > HROWS - 1) p = HROWS - 1;             // clamp (dup compute, no store)
    const int b  = (int)(p / HN);
    const int pp = (int)(p % HN);
    const int py = pp / HHP;
    const int px = pp % HHP;
    const float* xbase = x + (size_t)b * XBATCH + (size_t)py * HW + px;

    v8f acc = {};
    for (int t = 0; t < KITERS; ++t) {
        const int k0 = t * 4 + half * 2;          // even -> sOff pair is b64
        v2f a, bm;
        const int o0 = sOff[k0];
        const int o1 = sOff[k0 + 1];
        a[0] = xbase[o0];
        a[1] = xbase[o1];
        bm[0] = (n < 3) ? sW[k0 * 4 + n]       : 0.0f;
        bm[1] = (n < 3) ? sW[(k0 + 1) * 4 + n] : 0.0f;
        acc = __builtin_amdgcn_wmma_f32_16x16x4_f32(
                  false, a, false, bm, (short)0, acc, false, false);
    }

    // D layout: this lane holds column N=n, rows M = r + 8*half for acc[r].
    const long pbase = (long)blockIdx.x * 16 + 8 * half;
    if (n == 0) {
        const float bias = bq[0];
        #pragma unroll
        for (int r = 0; r < 8; ++r) {
            long pi = pbase + r;
            if (pi < HROWS) qo[pi] = acc[r] + bias;
        }
    } else if (n == 1) {
        const float bias = bk[0];
        #pragma unroll
        for (int r = 0; r < 8; ++r) {
            long pi = pbase + r;
            if (pi < HROWS) ko[pi] = acc[r] + bias;
        }
    } else if (n == 2) {
        const float bias = bv[0];
        #pragma unroll
        for (int r = 0; r < 8; ++r) {
            long pi = pbase + r;
            if (pi < HROWS) vo[pi] = acc[r] + bias;
        }
    }
}

// ---------------------------------------------------------------------------
// Kernel 2: rank-1 attention, exact streaming softmax.
//   sim[i,j] = q_i * k_j  =>  rowmax_i = q_i * (q_i>=0 ? kmax : kmin).
//   Each block stages k,v of one batch in LDS (64.8KB), reduces kmin/kmax,
//   then each thread owns one row i and scans all 8100 j with v_exp_f32.
// ---------------------------------------------------------------------------
__global__ void __launch_bounds__(256)
attn_rank1(const float* __restrict__ q,
           const float* __restrict__ k,
           const float* __restrict__ v,
           float* __restrict__ ctx)
{
    __shared__ float sk[HN];
    __shared__ float sv[HN];
    __shared__ float redmax[8];
    __shared__ float redmin[8];

    const int tid   = threadIdx.x;
    const int b     = blockIdx.x >> 5;   // 32 row-chunks per batch
    const int chunk = blockIdx.x & 31;

    float lmax = -3.402823466e38f;
    float lmin =  3.402823466e38f;
    for (int j = tid; j < HN; j += 256) {
        float kj = k[b * HN + j];
        sk[j] = kj;
        sv[j] = v[b * HN + j];
        lmax = fmaxf(lmax, kj);
        lmin = fminf(lmin, kj);
    }
    // wave-level reduce then cross-wave via tiny LDS array
    #pragma unroll
    for (int o = 16; o > 0; o >>= 1) {
        lmax = fmaxf(lmax, __shfl_xor(lmax, o, 32));
        lmin = fminf(lmin, __shfl_xor(lmin, o, 32));
    }
    if ((tid & 31) == 0) {
        redmax[tid >> 5] = lmax;
        redmin[tid >> 5] = lmin;
    }
    __syncthreads();
    float kmax = redmax[0], kmin = redmin[0];
    #pragma unroll
    for (int w = 1; w < 8; ++w) {
        kmax = fmaxf(kmax, redmax[w]);
        kmin = fminf(kmin, redmin[w]);
    }

    const int i = chunk * 256 + tid;
    if (i >= HN) return;

    const float a = q[b * HN + i];
    const float rowmax = (a >= 0.0f) ? a * kmax : a * kmin;
    float den = 0.0f, num = 0.0f;
    const float L2E = 1.44269504088896340736f;
    for (int j = 0; j < HN; ++j) {
        float t = fmaf(a, sk[j], -rowmax);
        float e = __builtin_amdgcn_exp2f(t * L2E);   // v_exp_f32
        den += e;
        num = fmaf(e, sv[j], num);
    }
    ctx[b * HN + i] = num / den;
}

// ---------------------------------------------------------------------------
// Kernel 3: 1x1 conv broadcast over C + half-pixel bilinear 90->96 + residual.
//   Purely bandwidth bound: read x, write out (4.7MB each). ctx map is L2-hot.
// ---------------------------------------------------------------------------
__global__ void __launch_bounds__(256)
out_upsample_residual(const float* __restrict__ x,
                      const float* __restrict__ ctx,
                      const float* __restrict__ w_out,
                      const float* __restrict__ b_out,
                      float* __restrict__ out)
{
    int idx = blockIdx.x * 256 + threadIdx.x;
    if (idx >= XTOTAL) return;

    int xc = idx % HW;
    int t  = idx / HW;
    int y  = t % HH;  t /= HH;
    int c  = t % HC;
    int b  = t / HC;

    const float scale = 90.0f / 96.0f;   // 0.9375 exact
    float sy = (y  + 0.5f) * scale - 0.5f;
    float sx = (xc + 0.5f) * scale - 0.5f;
    int y0 = (int)floorf(sy);
    int x0 = (int)floorf(sx);
    float fy = sy - (float)y0;
    float fx = sx - (float)x0;
    int y0c = max(y0, 0),     x0c = max(x0, 0);
    int y1c = min(y0 + 1, HHP - 1), x1c = min(x0 + 1, HHP - 1);

    const float* cm = ctx + (size_t)b * HN;
    float v00 = cm[y0c * HHP + x0c];
    float v01 = cm[y0c * HHP + x1c];
    float v10 = cm[y1c * HHP + x0c];
    float v11 = cm[y1c * HHP + x1c];
    float up = (1.0f - fy) * ((1.0f - fx) * v00 + fx * v01)
             +         fy  * ((1.0f - fx) * v10 + fx * v11);

    out[idx] = up * w_out[c] + b_out[c] + x[idx];
}

// ---------------------------------------------------------------------------
extern "C" void kernel_launch(void* const* d_in, const int* in_sizes, int n_in,
                              void* d_out, int out_size, void* d_ws, size_t ws_size,
                              hipStream_t stream)
{
    const float* x     = (const float*)d_in[0];
    const float* wq    = (const float*)d_in[1];
    const float* bq    = (const float*)d_in[2];
    const float* wk    = (const float*)d_in[3];
    const float* bk    = (const float*)d_in[4];
    const float* wv    = (const float*)d_in[5];
    const float* bv    = (const float*)d_in[6];
    const float* w_out = (const float*)d_in[7];
    const float* b_out = (const float*)d_in[8];
    float* out = (float*)d_out;

    float* ws  = (float*)d_ws;
    float* qa  = ws;
    float* ka  = ws + HROWS;
    float* va  = ws + 2 * HROWS;
    float* ctx = ws + 3 * HROWS;

    // 16200 rows / 16-pixel tiles = 1013 waves
    qkv_conv_wmma<<<(HROWS + 15) / 16, 32, 0, stream>>>(
        x, wq, wk, wv, bq, bk, bv, qa, ka, va);

    // 2 batches * 32 chunks of 256 rows
    attn_rank1<<<64, 256, 0, stream>>>(qa, ka, va, ctx);

    // 1,179,648 output elements
    out_upsample_residual<<<XTOTAL / 256, 256, 0, stream>>>(
        x, ctx, w_out, b_out, out);
}